// FraudGNN_57131654972028
// MI455X (gfx1250) — compile-verified
//
#include <hip/hip_runtime.h>
#include <hip/hip_bf16.h>

typedef __attribute__((ext_vector_type(2))) float v2f;
typedef __attribute__((ext_vector_type(8))) float v8f;
typedef int vi4 __attribute__((vector_size(16)));
typedef __attribute__((address_space(1))) vi4* gv4i_t;   // global int4*
typedef __attribute__((address_space(3))) vi4* lv4i_t;   // LDS int4*

#define N_NODES 50000
#define IN_DIM  128
#define N_EDGES 800000

// gfx1250 async global->LDS path (probe-guarded; falls back to synchronous staging)
#if defined(__has_builtin)
#  if __has_builtin(__builtin_amdgcn_global_load_async_to_lds_b128) && \
      __has_builtin(__builtin_amdgcn_s_wait_asynccnt)
#    define HAVE_ASYNC_LDS 1
#  endif
#endif
#ifndef HAVE_ASYNC_LDS
#  define HAVE_ASYNC_LDS 0
#endif

// ---------------------------------------------------------------- zero
__global__ __launch_bounds__(256) void zero_kernel(float4* __restrict__ p, int n4) {
    int i = blockIdx.x * blockDim.x + threadIdx.x;
    if (i < n4) p[i] = make_float4(0.f, 0.f, 0.f, 0.f);
}

// ---------------------------------------------------------------- degree: cnt[dst] += 1
__global__ __launch_bounds__(256) void degree_kernel(const long long* __restrict__ dst,
                                                     float* __restrict__ cnt, int E) {
    int e = blockIdx.x * blockDim.x + threadIdx.x;
    if (e < E) {
        int d = (int)dst[e];
        __hip_atomic_fetch_add(&cnt[d], 1.0f, __ATOMIC_RELAXED, __HIP_MEMORY_SCOPE_AGENT);
    }
}

// ---------------------------------------------------------------- scatter: agg[dst] += feat[src]
// one wave32 per edge; lane handles 4 contiguous floats (float4 gather + 4 f32 atomics,
// resolved in L2 since agg (25.6MB) fits the 192MB L2)
__global__ __launch_bounds__(256) void scatter_kernel(const float* __restrict__ feat,
                                                      const long long* __restrict__ src,
                                                      const long long* __restrict__ dst,
                                                      float* __restrict__ agg, int E) {
    int lane = threadIdx.x & 31;
    int e = blockIdx.x * 8 + (threadIdx.x >> 5);
    if (e >= E) return;
    int s = (int)src[e];
    int d = (int)dst[e];
    float4 v = *((const float4*)(feat + (size_t)s * IN_DIM) + lane);
    float* op = agg + (size_t)d * IN_DIM + lane * 4;
    __hip_atomic_fetch_add(op + 0, v.x, __ATOMIC_RELAXED, __HIP_MEMORY_SCOPE_AGENT);
    __hip_atomic_fetch_add(op + 1, v.y, __ATOMIC_RELAXED, __HIP_MEMORY_SCOPE_AGENT);
    __hip_atomic_fetch_add(op + 2, v.z, __ATOMIC_RELAXED, __HIP_MEMORY_SCOPE_AGENT);
    __hip_atomic_fetch_add(op + 3, v.w, __ATOMIC_RELAXED, __HIP_MEMORY_SCOPE_AGENT);
}

// ---------------------------------------------------------------- fused SAGE layer GEMM
// out = relu( (agg * 1/max(cnt,1)) @ Wl + xin @ Wr + b )
// A fragment (16x4 f32, v2f): lane<16 -> K = 4kt+{0,1}, lane>=16 -> K = 4kt+{2,3}; M = lane&15
// B fragment (4x16 f32, v2f): v0/v1 = rows k,k+1 (lanes 0-15), k+2,k+3 (lanes 16-31); N = lane&15
// C/D (16x16 f32, v8f): vgpr j -> M = (lane>>4)*8 + j, N = lane&15
__device__ inline void gemm_accum(const float* __restrict__ A, float scale,
                                  const float* __restrict__ sW,
                                  int r, int lane, v8f acc[4]) {
    const int khalf = (lane >> 4) * 2;   // 0 or 2
    const int col   = lane & 15;
    for (int kt = 0; kt < 32; ++kt) {
        int k = kt * 4 + khalf;
        float2 av = *(const float2*)(A + (size_t)r * IN_DIM + k);
        v2f a;
        a.x = av.x * scale;
        a.y = av.y * scale;
#pragma unroll
        for (int nt = 0; nt < 4; ++nt) {
            v2f b;
            b.x = sW[(k + 0) * 64 + nt * 16 + col];
            b.y = sW[(k + 1) * 64 + nt * 16 + col];
            acc[nt] = __builtin_amdgcn_wmma_f32_16x16x4_f32(
                false, a, false, b, (short)0, acc[nt], false, false);
        }
    }
}

__global__ __launch_bounds__(256) void sage_layer_kernel(const float* __restrict__ agg,
                                                         const float* __restrict__ cnt,
                                                         const float* __restrict__ xin,
                                                         const float* __restrict__ Wl,
                                                         const float* __restrict__ Wr,
                                                         const float* __restrict__ bias,
                                                         float* __restrict__ out) {
    __shared__ float sWl[128 * 64];
    __shared__ float sWr[128 * 64];

    const int colbase = blockIdx.y * 64;

#if HAVE_ASYNC_LDS
    // stage both 128x64 weight slabs with async global->LDS b128 copies
    // (16B global chunk -> 16B LDS chunk, 1:1 in this slab layout)
    for (int i = threadIdx.x; i < 4096; i += 256) {
        int m  = i >> 11;               // 0 -> Wl, 1 -> Wr
        int q  = i & 2047;
        int k  = q >> 4;                // weight row 0..127
        int c4 = (q & 15) * 4;          // slab column 0..60 step 4
        float* g = const_cast<float*>((m ? Wr : Wl) + k * 128 + colbase + c4);
        float* l = (m ? sWr : sWl) + k * 64 + c4;
        __builtin_amdgcn_global_load_async_to_lds_b128((gv4i_t)g, (lv4i_t)l, 0, 0);
    }
    __builtin_amdgcn_s_wait_asynccnt(0);
    __syncthreads();
#else
    for (int i = threadIdx.x; i < 128 * 64; i += 256) {
        int k = i >> 6, c = i & 63;
        sWl[i] = Wl[k * 128 + colbase + c];
        sWr[i] = Wr[k * 128 + colbase + c];
    }
    __syncthreads();
#endif

    const int lane  = threadIdx.x & 31;
    const int strip = blockIdx.x * 8 + (threadIdx.x >> 5);   // 16-row strip index
    if (strip >= N_NODES / 16) return;                        // wave-uniform: EXEC stays all-1s
    const int row0 = strip * 16;
    const int col  = lane & 15;
    const int r    = row0 + col;                              // this lane's A row (fixed)

    const float c  = cnt[r];
    const float invc = 1.0f / fmaxf(c, 1.0f);

    v8f acc[4];
#pragma unroll
    for (int nt = 0; nt < 4; ++nt) {
        float b = bias[colbase + nt * 16 + col];
#pragma unroll
        for (int j = 0; j < 8; ++j) acc[nt][j] = b;
    }

    gemm_accum(agg, invc, sWl, r, lane, acc);   // mean-neighbor term
    gemm_accum(xin, 1.0f, sWr, r, lane, acc);   // self term

    const int mbase = (lane >> 4) * 8;
#pragma unroll
    for (int nt = 0; nt < 4; ++nt) {
        int cc = colbase + nt * 16 + col;
#pragma unroll
        for (int j = 0; j < 8; ++j) {
            out[(size_t)(row0 + mbase + j) * IN_DIM + cc] = fmaxf(acc[nt][j], 0.f);
        }
    }
}

// ---------------------------------------------------------------- classifier + log_softmax
__global__ __launch_bounds__(256) void classify_kernel(const float* __restrict__ h,
                                                       const float* __restrict__ Wc,
                                                       const float* __restrict__ bc,
                                                       float* __restrict__ out, int N) {
    int lane = threadIdx.x & 31;
    int node = blockIdx.x * 8 + (threadIdx.x >> 5);
    if (node >= N) return;
    float4 v = *((const float4*)(h + (size_t)node * IN_DIM) + lane);
    int d0 = lane * 4;
    float z0 = v.x * Wc[(d0 + 0) * 2 + 0] + v.y * Wc[(d0 + 1) * 2 + 0] +
               v.z * Wc[(d0 + 2) * 2 + 0] + v.w * Wc[(d0 + 3) * 2 + 0];
    float z1 = v.x * Wc[(d0 + 0) * 2 + 1] + v.y * Wc[(d0 + 1) * 2 + 1] +
               v.z * Wc[(d0 + 2) * 2 + 1] + v.w * Wc[(d0 + 3) * 2 + 1];
#pragma unroll
    for (int off = 16; off > 0; off >>= 1) {
        z0 += __shfl_xor(z0, off, 32);
        z1 += __shfl_xor(z1, off, 32);
    }
    if (lane == 0) {
        z0 += bc[0];
        z1 += bc[1];
        float m = fmaxf(z0, z1);
        float lse = m + logf(expf(z0 - m) + expf(z1 - m));
        out[(size_t)node * 2 + 0] = z0 - lse;
        out[(size_t)node * 2 + 1] = z1 - lse;
    }
}

// ---------------------------------------------------------------- launch
extern "C" void kernel_launch(void* const* d_in, const int* in_sizes, int n_in,
                              void* d_out, int out_size, void* d_ws, size_t ws_size,
                              hipStream_t stream) {
    const float*      x   = (const float*)d_in[0];
    const long long*  ei  = (const long long*)d_in[1];   // int64 edge_index [2, E]
    const float*      W1l = (const float*)d_in[2];
    const float*      W1r = (const float*)d_in[3];
    const float*      b1  = (const float*)d_in[4];
    const float*      W2l = (const float*)d_in[5];
    const float*      W2r = (const float*)d_in[6];
    const float*      b2  = (const float*)d_in[7];
    const float*      Wc  = (const float*)d_in[8];
    const float*      bc  = (const float*)d_in[9];
    float*            out = (float*)d_out;

    const int N = N_NODES, E = N_EDGES;
    const long long* src = ei;
    const long long* dst = ei + E;

    // workspace layout (floats): cnt[50048] | agg[N*128] | h1[N*128] | h2[N*128]  (~77 MB)
    float* ws  = (float*)d_ws;
    float* cnt = ws;
    float* agg = ws + 50048;
    float* h1  = agg + (size_t)N * IN_DIM;
    float* h2  = h1 + (size_t)N * IN_DIM;

    const int strips = N / 16;                       // 3125 (exact)
    dim3 ggrid((strips + 7) / 8, 2);                 // 8 waves/block, 2 column halves

    // ---- layer 1
    {
        int n4 = (50048 + N * IN_DIM) / 4;
        zero_kernel<<<(n4 + 255) / 256, 256, 0, stream>>>((float4*)cnt, n4);
    }
    degree_kernel<<<(E + 255) / 256, 256, 0, stream>>>(dst, cnt, E);
    scatter_kernel<<<(E + 7) / 8, 256, 0, stream>>>(x, src, dst, agg, E);
    sage_layer_kernel<<<ggrid, 256, 0, stream>>>(agg, cnt, x, W1l, W1r, b1, h1);

    // ---- layer 2 (reuse degree counts)
    {
        int n4 = (N * IN_DIM) / 4;
        zero_kernel<<<(n4 + 255) / 256, 256, 0, stream>>>((float4*)agg, n4);
    }
    scatter_kernel<<<(E + 7) / 8, 256, 0, stream>>>(h1, src, dst, agg, E);
    sage_layer_kernel<<<ggrid, 256, 0, stream>>>(agg, cnt, h1, W2l, W2r, b2, h2);

    // ---- classifier + log_softmax
    classify_kernel<<<(N + 7) / 8, 256, 0, stream>>>(h2, Wc, bc, out, N);
}